// HandwrittenFastAdderTransformer_48017734369364
// MI455X (gfx1250) — compile-verified
//
#include <hip/hip_runtime.h>
#include <math.h>

typedef __attribute__((ext_vector_type(2))) float v2f;
typedef __attribute__((ext_vector_type(8))) float v8f;

#define BATCH 16
#define SEQ   1024
#define NH    4
#define VOCABN 10

// workspace layout (float offsets)
#define XBUF_OFF 0                              // [B*T][4]   x (3 + pad)
#define QBUF_OFF (XBUF_OFF + BATCH*SEQ*4)       // [B*T][8]   q (4 heads x 2), pre-scaled by 1/sqrt(2)
#define KBUF_OFF (QBUF_OFF + BATCH*SEQ*8)       // [B*T][2]   k
#define VBUF_OFF (KBUF_OFF + BATCH*SEQ*2)       // [B*T][2]   v
#define CTX_OFF  (VBUF_OFF + BATCH*SEQ*2)       // [B][4][T][2] attention context

// ---------------- Phase 1: embed -> rmsnorm -> q/k/v (+qk-norm, rope) ----------------
__global__ void p1_embed_qkv(const int* __restrict__ tok,
                             const float* __restrict__ embed,
                             const float* __restrict__ in_nw,
                             const float* __restrict__ q_nw,
                             const float* __restrict__ k_nw,
                             const float* __restrict__ k_pw,
                             const float* __restrict__ v_pw,
                             const float* __restrict__ q_pw,
                             float* __restrict__ ws) {
  int idx = blockIdx.x * blockDim.x + threadIdx.x;   // < BATCH*SEQ
  int t = idx & (SEQ - 1);
  int d = tok[idx];
  float x0 = embed[d*3+0], x1 = embed[d*3+1], x2 = embed[d*3+2];
  float var = (x0*x0 + x1*x1 + x2*x2) * (1.0f/3.0f);
  float rn  = rsqrtf(var + 1e-6f);
  float y0 = x0*rn*in_nw[0], y1 = x1*rn*in_nw[1], y2 = x2*rn*in_nw[2];

  float* xb = ws + XBUF_OFF + idx*4;
  xb[0] = x0; xb[1] = x1; xb[2] = x2; xb[3] = 0.0f;

  float sn, cs;
  __sincosf((float)t, &sn, &cs);

  const float inv_sqrt_hd = 0.70710678118654752440f;   // 1/sqrt(head_dim)
  float* qb = ws + QBUF_OFF + idx*8;
#pragma unroll
  for (int h = 0; h < 4; ++h) {
    float qa = q_pw[(2*h+0)*3+0]*y0 + q_pw[(2*h+0)*3+1]*y1 + q_pw[(2*h+0)*3+2]*y2;
    float qc = q_pw[(2*h+1)*3+0]*y0 + q_pw[(2*h+1)*3+1]*y1 + q_pw[(2*h+1)*3+2]*y2;
    float qv = (qa*qa + qc*qc) * 0.5f;
    float qr = rsqrtf(qv + 1e-6f);
    float na = qa*qr*q_nw[0], nb = qc*qr*q_nw[1];
    float ra = na*cs - nb*sn;
    float rb = na*sn + nb*cs;
    qb[2*h+0] = ra * inv_sqrt_hd;
    qb[2*h+1] = rb * inv_sqrt_hd;
  }

  float ka = k_pw[0]*y0 + k_pw[1]*y1 + k_pw[2]*y2;
  float kc = k_pw[3]*y0 + k_pw[4]*y1 + k_pw[5]*y2;
  float kv = (ka*ka + kc*kc) * 0.5f;
  float kr = rsqrtf(kv + 1e-6f);
  float na = ka*kr*k_nw[0], nb = kc*kr*k_nw[1];
  float* kb = ws + KBUF_OFF + idx*2;
  kb[0] = na*cs - nb*sn;
  kb[1] = na*sn + nb*cs;

  float* vb = ws + VBUF_OFF + idx*2;
  vb[0] = v_pw[0]*y0 + v_pw[1]*y1 + v_pw[2]*y2;
  vb[1] = v_pw[3]*y0 + v_pw[4]*y1 + v_pw[5]*y2;
}

// ---------------- Phase 2: causal flash attention via WMMA 16x16x4 f32 ----------------
// One wave per (batch, head, 16-query block). D[m][n] = k_m . q_n.
// In C/D layout query column n lives in lanes n (rows 0-7) and n+16 (rows 8-15):
// each lane keeps its own online-softmax state, merged once at the end via shfl_xor(16).
__global__ void p2_attention(float* __restrict__ ws) {
  int bid  = blockIdx.x;
  int qb   = bid & 63;          // query block (16 queries)
  int h    = (bid >> 6) & 3;
  int b    = bid >> 8;
  int lane = threadIdx.x;
  int n    = lane & 15;
  int half = lane >> 4;
  int q0   = qb * 16;
  int base = half * 8;          // row offset this lane owns in D

  const float* qbuf = ws + QBUF_OFF + (size_t)(b*SEQ)*8;
  const float* kbuf = ws + KBUF_OFF + (size_t)(b*SEQ)*2;
  const float* vbuf = ws + VBUF_OFF + (size_t)(b*SEQ)*2;

  // B operand: B[k][n] = q[q0+n][k]; lanes 16-31 carry K rows 2,3 = 0
  v2f bq;
  if (half == 0) { bq.x = qbuf[(q0+n)*8 + h*2+0]; bq.y = qbuf[(q0+n)*8 + h*2+1]; }
  else           { bq.x = 0.0f; bq.y = 0.0f; }

  float m = -__builtin_inff(), lsum = 0.0f, a0 = 0.0f, a1 = 0.0f;

  for (int kt = 0; kt <= qb; ++kt) {
    int k0 = kt * 16;
    // A operand: A[m][k] = k[k0+m][k]; lanes 16-31 carry K cols 2,3 = 0
    v2f av;
    if (half == 0) { av.x = kbuf[(k0+n)*2+0]; av.y = kbuf[(k0+n)*2+1]; }
    else           { av.x = 0.0f; av.y = 0.0f; }

    v8f dacc = {};
    dacc = __builtin_amdgcn_wmma_f32_16x16x4_f32(false, av, false, bq,
                                                 (short)0, dacc, false, false);
    if (kt == qb) {           // diagonal tile: causal mask (key row > query col)
#pragma unroll
      for (int r = 0; r < 8; ++r)
        dacc[r] = (base + r <= n) ? dacc[r] : -__builtin_inff();
    }

    float tm = fmaxf(fmaxf(fmaxf(dacc[0], dacc[1]), fmaxf(dacc[2], dacc[3])),
                     fmaxf(fmaxf(dacc[4], dacc[5]), fmaxf(dacc[6], dacc[7])));
    float mn = fmaxf(m, tm);
    if (mn > -3.0e38f) {      // skip lanes whose whole tile is masked
      float alpha = __expf(m - mn);
      float ps = 0.0f, p0 = 0.0f, p1 = 0.0f;
#pragma unroll
      for (int r = 0; r < 8; ++r) {
        float p  = __expf(dacc[r] - mn);
        float v0 = vbuf[(k0 + base + r)*2 + 0];
        float v1 = vbuf[(k0 + base + r)*2 + 1];
        ps += p; p0 += p*v0; p1 += p*v1;
      }
      lsum = lsum*alpha + ps;
      a0   = a0*alpha   + p0;
      a1   = a1*alpha   + p1;
      m    = mn;
    }
  }

  // merge the two half-wave partial softmaxes for each query column
  float om  = __shfl_xor(m,    16, 32);
  float ol  = __shfl_xor(lsum, 16, 32);
  float oa0 = __shfl_xor(a0,   16, 32);
  float oa1 = __shfl_xor(a1,   16, 32);
  float M   = fmaxf(m, om);
  float s1  = __expf(m - M), s2 = __expf(om - M);
  float L   = lsum*s1 + ol*s2;
  float inv = 1.0f / L;
  float c0  = (a0*s1 + oa0*s2) * inv;
  float c1  = (a1*s1 + oa1*s2) * inv;

  if (half == 0) {
    float* ctx = ws + CTX_OFF + (size_t)(((b*NH + h)*SEQ) + (q0 + n)) * 2;
    ctx[0] = c0; ctx[1] = c1;
  }
}

// ---------------- Phase 3: o_proj + MLP + final norm + WMMA logits ----------------
// One wave per 16 positions; both half-waves compute the same 16 positions so that
// lane l>=16 can supply the K=2 component of the A operand without shuffles.
__global__ void p3_tail_logits(const float* __restrict__ ws,
                               const float* __restrict__ embed,
                               const float* __restrict__ post_nw,
                               const float* __restrict__ final_nw,
                               const float* __restrict__ o_pw,
                               const float* __restrict__ gate_pw,
                               const float* __restrict__ up_pw,
                               const float* __restrict__ down_pw,
                               float* __restrict__ out) {
  int tile = blockIdx.x;
  int lane = threadIdx.x;
  int n    = lane & 15;
  int half = lane >> 4;
  int idx  = tile*16 + n;                  // position index < BATCH*SEQ
  int b    = idx >> 10, t = idx & (SEQ - 1);

  const float* xb = ws + XBUF_OFF + idx*4;
  float x0 = xb[0], x1 = xb[1], x2 = xb[2];

  float ctx[8];
#pragma unroll
  for (int h = 0; h < 4; ++h) {
    const float* cp = ws + CTX_OFF + (size_t)(((b*NH + h)*SEQ) + t) * 2;
    ctx[2*h+0] = cp[0]; ctx[2*h+1] = cp[1];
  }
  float ao0 = 0.0f, ao1 = 0.0f, ao2 = 0.0f;
#pragma unroll
  for (int i = 0; i < 8; ++i) {
    ao0 += o_pw[0*8+i]*ctx[i];
    ao1 += o_pw[1*8+i]*ctx[i];
    ao2 += o_pw[2*8+i]*ctx[i];
  }
  x0 += ao0; x1 += ao1; x2 += ao2;

  float var = (x0*x0 + x1*x1 + x2*x2) * (1.0f/3.0f);
  float rn  = rsqrtf(var + 1e-6f);
  float y0 = x0*rn*post_nw[0], y1 = x1*rn*post_nw[1], y2 = x2*rn*post_nw[2];

  float m0 = 0.0f, m1 = 0.0f, m2 = 0.0f;
#pragma unroll
  for (int i = 0; i < 4; ++i) {
    float g = gate_pw[i*3+0]*y0 + gate_pw[i*3+1]*y1 + gate_pw[i*3+2]*y2;
    float u = up_pw[i*3+0]*y0 + up_pw[i*3+1]*y1 + up_pw[i*3+2]*y2;
    float act = (g / (1.0f + __expf(-g))) * u;     // silu(g) * u
    m0 += down_pw[0*4+i]*act;
    m1 += down_pw[1*4+i]*act;
    m2 += down_pw[2*4+i]*act;
  }
  x0 += m0; x1 += m1; x2 += m2;

  var = (x0*x0 + x1*x1 + x2*x2) * (1.0f/3.0f);
  rn  = rsqrtf(var + 1e-6f);
  float z0 = x0*rn*final_nw[0], z1 = x1*rn*final_nw[1], z2 = x2*rn*final_nw[2];

  // logits[m][n] = sum_k x_out[m][k] * embed[n][k], K padded 3->4
  v2f av, bv;
  if (half == 0) { av.x = z0; av.y = z1; }
  else           { av.x = z2; av.y = 0.0f; }
  if (n < VOCABN) {
    if (half == 0) { bv.x = embed[n*3+0]; bv.y = embed[n*3+1]; }
    else           { bv.x = embed[n*3+2]; bv.y = 0.0f; }
  } else { bv.x = 0.0f; bv.y = 0.0f; }

  v8f dacc = {};
  dacc = __builtin_amdgcn_wmma_f32_16x16x4_f32(false, av, false, bv,
                                               (short)0, dacc, false, false);
  if (n < VOCABN) {
#pragma unroll
    for (int r = 0; r < 8; ++r) {
      int row = tile*16 + half*8 + r;
      out[row*VOCABN + n] = dacc[r];
    }
  }
}

extern "C" void kernel_launch(void* const* d_in, const int* in_sizes, int n_in,
                              void* d_out, int out_size, void* d_ws, size_t ws_size,
                              hipStream_t stream) {
  const int*   tok     = (const int*)  d_in[0];   // token_ids [16*1024]
  const float* embed   = (const float*)d_in[1];   // embed_weight [10*3]
  const float* in_nw   = (const float*)d_in[2];   // input_norm_weight [3]
  const float* post_nw = (const float*)d_in[3];   // post_norm_weight [3]
  const float* fin_nw  = (const float*)d_in[4];   // final_norm_weight [3]
  const float* q_nw    = (const float*)d_in[5];   // q_norm_weight [2]
  const float* k_nw    = (const float*)d_in[6];   // k_norm_weight [2]
  const float* k_pw    = (const float*)d_in[7];   // k_proj_weight [2*3]
  const float* v_pw    = (const float*)d_in[8];   // v_proj_weight [2*3]
  const float* q_pw    = (const float*)d_in[9];   // q_proj_weight [8*3]
  const float* o_pw    = (const float*)d_in[10];  // o_proj_weight [3*8]
  const float* gate_pw = (const float*)d_in[11];  // gate_proj_weight [4*3]
  const float* up_pw   = (const float*)d_in[12];  // up_proj_weight [4*3]
  const float* down_pw = (const float*)d_in[13];  // down_proj_weight [3*4]
  float* ws  = (float*)d_ws;
  float* out = (float*)d_out;

  p1_embed_qkv<<<dim3((BATCH*SEQ)/256), dim3(256), 0, stream>>>(
      tok, embed, in_nw, q_nw, k_nw, k_pw, v_pw, q_pw, ws);

  p2_attention<<<dim3(BATCH*NH*(SEQ/16)), dim3(32), 0, stream>>>(ws);

  p3_tail_logits<<<dim3((BATCH*SEQ)/16), dim3(32), 0, stream>>>(
      ws, embed, post_nw, fin_nw, o_pw, gate_pw, up_pw, down_pw, out);
}